// BlockSAGE_78254304133462
// MI455X (gfx1250) — compile-verified
//
#include <hip/hip_runtime.h>
#include <hip/hip_bf16.h>

typedef __attribute__((ext_vector_type(2))) float v2f;
typedef __attribute__((ext_vector_type(8))) float v8f;

// ---------------------------------------------------------------------------
// Edge scatter: one wave (32 lanes) per edge. Gathers the source row
// (optionally through gidx indirection) and atomically accumulates it into
// agg[dst]. Lane 0 bumps the per-destination count.
// agg region is L2-resident (46 MB / 8 MB << 192 MB L2) so the f32 atomics
// resolve in L2 atomic units, not HBM. This phase is the roofline bottleneck:
// E0 * 512 B ~= 692 MB of row gathers from HBM (~30 us at 23.3 TB/s).
// ---------------------------------------------------------------------------
__global__ void __launch_bounds__(256)
scatter_mean(const float* __restrict__ X,     // source feature table [., F]
             const int*   __restrict__ gidx,  // optional row indirection (or null)
             const int*   __restrict__ esrc,  // [E]
             const int*   __restrict__ edst,  // [E]
             float* __restrict__ agg,         // [num_dst, F] (pre-zeroed)
             float* __restrict__ cnt,         // [num_dst]    (pre-zeroed)
             int E, int F)
{
    int e    = (int)((blockIdx.x * blockDim.x + threadIdx.x) >> 5);
    int lane = threadIdx.x & 31;
    if (e >= E) return;

    int s = esrc[e];
    int d = edst[e];
    int g = gidx ? gidx[s] : s;

    const float* row = X   + (long long)g * F;
    float*       out = agg + (long long)d * F;

    for (int f = lane * 4; f < F; f += 128) {
        float4 v = *(const float4*)(row + f);
        unsafeAtomicAdd(out + f + 0, v.x);
        unsafeAtomicAdd(out + f + 1, v.y);
        unsafeAtomicAdd(out + f + 2, v.z);
        unsafeAtomicAdd(out + f + 3, v.w);
    }
    if (lane == 0) unsafeAtomicAdd(cnt + d, 1.0f);
}

// ---------------------------------------------------------------------------
// agg[i] /= max(cnt[row], 1)   (elementwise over [M, F]); F = 1 << fshift.
// ---------------------------------------------------------------------------
__global__ void __launch_bounds__(256)
scale_mean(float* __restrict__ agg, const float* __restrict__ cnt,
           long long total, int fshift)
{
    long long i = (long long)blockIdx.x * blockDim.x + threadIdx.x;
    if (i >= total) return;
    float c = cnt[i >> fshift];
    c = c > 1.0f ? c : 1.0f;
    agg[i] = agg[i] / c;
}

// ---------------------------------------------------------------------------
// Fused dual GEMM on WMMA fp32:  C = act( A1@B1 + A2@B2 + bias )
//   A1 : [M,K] dense (the segment means)
//   A2 : gathered rows A2base[gidx2[m]] (or A2base[m] if gidx2==null), [.,K]
//   B1,B2 : [K,N] row-major weights
//
// One wave computes a (16*MSUB) x 16 tile of C with MSUB accumulators.
// Per k-step ALL operands (2*MSUB v2f A loads + 4 b32 B loads) are fetched
// into distinct locals first, THEN 2*MSUB v_wmma_f32_16x16x4_f32 issue
// back-to-back. This lets the backend clause the loads and cover their
// latency with the WMMA burst instead of ping-ponging load->wait0->wmma.
// K loop unrolled 2x for cross-iteration overlap.
//
// Operand layout (wave32, per ISA 7.12.2):
//   A 16x4 : lane L (L<16) holds A[M=L][K=k..k+1]; lanes 16..31 hold
//            K=k+2..k+3  -> per-lane float2 at (row = lane&15,
//            kk = k + 2*(lane>>4)).
//   B 4x16 : mirrored — per-lane float2 = B[kk][col], B[kk+1][col].
//   C/D    : VGPR r -> row (subtile_base + r + 8*(lane>>4)), col = tn+(lane&15).
// Early tile-out-of-range return is wave-uniform, so EXEC is all-ones for
// every executed WMMA (ISA requirement).
// ---------------------------------------------------------------------------
template <int MSUB>
__global__ void __launch_bounds__(256)
gemm_dual_wmma(const float* __restrict__ A1,
               const float* __restrict__ A2base,
               const int*   __restrict__ gidx2,
               const float* __restrict__ B1,
               const float* __restrict__ B2,
               const float* __restrict__ bias,
               float* __restrict__ C,
               int M, int K, int N, int doRelu)
{
    const int lane   = threadIdx.x & 31;
    const int wave   = threadIdx.x >> 5;
    const int tilesN = N >> 4;
    const int nTiles = (M / (16 * MSUB)) * tilesN;

    int tile = blockIdx.x * 8 + wave;       // wave-uniform
    if (tile >= nTiles) return;             // wave-uniform exit (EXEC stays full)

    const int tm  = (tile / tilesN) * (16 * MSUB);
    const int tn  = (tile % tilesN) << 4;
    const int r16 = lane & 15;
    const int hi  = lane >> 4;              // 0: K pair {k,k+1}, 1: {k+2,k+3}
    const int col = tn + r16;

    // Per-lane base pointers, pre-offset by this lane's K pair (hi*2 floats).
    const float* a1p[MSUB];
    const float* a2p[MSUB];
#pragma unroll
    for (int s = 0; s < MSUB; ++s) {
        const int m = tm + 16 * s + r16;
        a1p[s] = A1 + (long long)m * K + (hi << 1);
        const int g2 = gidx2 ? gidx2[m] : m;
        a2p[s] = A2base + (long long)g2 * K + (hi << 1);
    }
    const float* b1p = B1 + (long long)(hi << 1) * N + col;
    const float* b2p = B2 + (long long)(hi << 1) * N + col;

    v8f acc[MSUB];
#pragma unroll
    for (int s = 0; s < MSUB; ++s)
        acc[s] = (v8f){0.f, 0.f, 0.f, 0.f, 0.f, 0.f, 0.f, 0.f};

#pragma unroll 2
    for (int k = 0; k < K; k += 4) {
        // ---- issue ALL loads for this k-step first (distinct destinations)
        v2f b1v, b2v;
        b1v.x = b1p[(long long)k * N];
        b1v.y = b1p[(long long)(k + 1) * N];
        b2v.x = b2p[(long long)k * N];
        b2v.y = b2p[(long long)(k + 1) * N];

        v2f a1v[MSUB], a2v[MSUB];
#pragma unroll
        for (int s = 0; s < MSUB; ++s) {
            a1v[s] = *(const v2f*)(a1p[s] + k);
            a2v[s] = *(const v2f*)(a2p[s] + k);
        }

        // ---- then burst the WMMAs (mutually independent except acc chains)
#pragma unroll
        for (int s = 0; s < MSUB; ++s)
            acc[s] = __builtin_amdgcn_wmma_f32_16x16x4_f32(
                         false, a1v[s], false, b1v, (short)0, acc[s], false, false);
#pragma unroll
        for (int s = 0; s < MSUB; ++s)
            acc[s] = __builtin_amdgcn_wmma_f32_16x16x4_f32(
                         false, a2v[s], false, b2v, (short)0, acc[s], false, false);
    }

    const float bv = bias[col];
#pragma unroll
    for (int s = 0; s < MSUB; ++s) {
#pragma unroll
        for (int r = 0; r < 8; ++r) {
            float v = acc[s][r] + bv;
            if (doRelu) v = v > 0.f ? v : 0.f;
            const int m = tm + 16 * s + r + (hi << 3);
            C[(long long)m * N + col] = v;
        }
    }
}

// ---------------------------------------------------------------------------
// Host-side orchestration (graph-capture safe: only async APIs on `stream`).
// ---------------------------------------------------------------------------
extern "C" void kernel_launch(void* const* d_in, const int* in_sizes, int n_in,
                              void* d_out, int out_size, void* d_ws, size_t ws_size,
                              hipStream_t stream)
{
    (void)n_in; (void)ws_size;

    const float* x_all = (const float*)d_in[0];
    const int*   n_id  = (const int*)  d_in[1];
    const int*   esrc0 = (const int*)  d_in[2];
    const int*   edst0 = (const int*)  d_in[3];
    const int*   esrc1 = (const int*)  d_in[4];
    const int*   edst1 = (const int*)  d_in[5];
    const float* W_l0  = (const float*)d_in[6];
    const float* b_l0  = (const float*)d_in[7];
    const float* W_r0  = (const float*)d_in[8];
    const float* W_l1  = (const float*)d_in[9];
    const float* b_l1  = (const float*)d_in[10];
    const float* W_r1  = (const float*)d_in[11];

    const int F_IN = 128, HID = 256, OUT = 128;
    const int E0   = in_sizes[2];
    const int E1   = in_sizes[4];
    const int ND0  = 90112;                 // fixed problem geometry
    const int ND1  = out_size / OUT;        // 8192

    // Workspace carve-up (floats).
    float* agg0 = (float*)d_ws;                          // [ND0, 128]
    float* cnt0 = agg0 + (size_t)ND0 * F_IN;             // [ND0]
    float* h    = cnt0 + (size_t)ND0;                    // [ND0, 256]
    float* agg1 = h    + (size_t)ND0 * HID;              // [ND1, 256]
    float* cnt1 = agg1 + (size_t)ND1 * HID;              // [ND1]

    // Zero the accumulation regions (agg+cnt are contiguous per layer).
    hipMemsetAsync(agg0, 0, (size_t)ND0 * (F_IN + 1) * sizeof(float), stream);
    hipMemsetAsync(agg1, 0, (size_t)ND1 * (HID  + 1) * sizeof(float), stream);

    // ---- Layer 0 ----
    {
        int blocks = (E0 + 7) / 8;                       // 8 waves/block, 1 edge/wave
        scatter_mean<<<blocks, 256, 0, stream>>>(
            x_all, n_id, esrc0, edst0, agg0, cnt0, E0, F_IN);

        long long tot = (long long)ND0 * F_IN;
        scale_mean<<<(int)((tot + 255) / 256), 256, 0, stream>>>(
            agg0, cnt0, tot, /*fshift=*/7);

        // 64x16 tiles: (90112/64) * (256/16) = 22528 waves
        int nTiles = (ND0 / 64) * (HID / 16);
        gemm_dual_wmma<4><<<(nTiles + 7) / 8, 256, 0, stream>>>(
            agg0, x_all, n_id, W_l0, W_r0, b_l0, h,
            ND0, F_IN, HID, /*relu=*/1);
    }

    // ---- Layer 1 ----
    {
        int blocks = (E1 + 7) / 8;
        scatter_mean<<<blocks, 256, 0, stream>>>(
            h, nullptr, esrc1, edst1, agg1, cnt1, E1, HID);

        long long tot = (long long)ND1 * HID;
        scale_mean<<<(int)((tot + 255) / 256), 256, 0, stream>>>(
            agg1, cnt1, tot, /*fshift=*/8);

        // 32x16 tiles: (8192/32) * (128/16) = 2048 waves (keeps SIMDs fed)
        int nTiles = (ND1 / 32) * (OUT / 16);
        gemm_dual_wmma<2><<<(nTiles + 7) / 8, 256, 0, stream>>>(
            agg1, h, nullptr, W_l1, W_r1, b_l1, (float*)d_out,
            ND1, HID, OUT, /*relu=*/0);
    }
}